// PositionalTopDownHTMM_83623013253132
// MI455X (gfx1250) — compile-verified
//
#include <hip/hip_runtime.h>
#include <math.h>

// MI455X / gfx1250: wave32, WMMA 16x16xK. V_WMMA_F32_16X16X4_F32 (full f32):
// problem is bandwidth/latency bound (~0.7 GFLOP, ~120 MB of intermediates
// that fit in the 192 MB L2), so full precision is free and the log-
// likelihood recursion (4-way products over 8 levels feeding log()) needs it.
//
// Per-node layout is [node][g][s] (s = hidden state, 32 floats/node) so the
// WMMA K-operand pairs (s, s+1) are contiguous -> global_load_b64, and the
// down-pass D stores are lane-coalesced.

typedef float v2f __attribute__((ext_vector_type(2)));
typedef float v8f __attribute__((ext_vector_type(8)));
typedef unsigned int u32x4 __attribute__((ext_vector_type(4)));
typedef int i32x8 __attribute__((ext_vector_type(8)));
typedef int i32x4 __attribute__((ext_vector_type(4)));

#define MEMIT 100

// Workspace layout (float offsets)
#define WS_SMA_DC 0                       // [j][g][d][c] = P(c|d,j,g), 1024 floats
#define WS_SMA_CD 1024                    // [j][g][c][d], 1024 floats
#define WS_SMBT   2048                    // [m][g][c], 3200 floats
#define WS_PRIOR  8192                    // [node][g][s], DIM*32 floats
#define DIMTOT    699048
#define WS_BETA0  (WS_PRIOR + DIMTOT*32)  // 22377728
#define WS_BETA1  (WS_BETA0 + 131072*32)  // 26572032
// total ws: 30,766,336 floats = ~123.1 MB

#if defined(__has_builtin)
#if __has_builtin(__builtin_amdgcn_tensor_load_to_lds) && __has_builtin(__builtin_amdgcn_s_wait_tensorcnt)
#define USE_TDM 1
#endif
#endif

// Preload a 4KB (1024-float) table from ws+off into LDS via the Tensor Data
// Mover (one DMA per block) when available; plain loads otherwise.
__device__ static inline void load_table_lds(float* sA, const float* __restrict__ ws, int off) {
#ifdef USE_TDM
  if (threadIdx.x == 0) {
    unsigned lds = (unsigned)(unsigned long long)(uintptr_t)sA;   // low 32b = LDS offset (ISA 10.2)
    unsigned long long ga = (unsigned long long)(uintptr_t)(ws + off);
    u32x4 g0 = { 1u,                                             // count=1, user mode
                 lds,                                            // lds_addr
                 (unsigned)(ga & 0xffffffffu),                   // global_addr[31:0]
                 (unsigned)((ga >> 32) & 0x01ffffffu) | 0x80000000u }; // addr[56:32] | type=2
    i32x8 g1 = { 0x20000,                 // data_size = 4B
                 0x04000000,              // tensor_dim0 = 1024 (bits 79:48, low half)
                 0x00010000,              // tensor_dim0 hi=0 | tensor_dim1 = 1
                 0x04000000,              // tensor_dim1 hi=0 | tile_dim0 = 1024
                 1,                       // tile_dim1 = 1, tile_dim2 = 0
                 1024,                    // tensor_dim0_stride lo
                 0x04000000,              // stride0 hi=0 | tensor_dim1_stride lo = 1024
                 0 };
    i32x4 z4 = {0, 0, 0, 0};
    i32x8 z8 = {0, 0, 0, 0, 0, 0, 0, 0};
    __builtin_amdgcn_tensor_load_to_lds(g0, g1, z4, z4, z8, 0);  // 6-arg form (clang-23)
  }
  __builtin_amdgcn_s_wait_tensorcnt(0);
  __syncthreads();
#else
  for (int i = threadIdx.x; i < 1024; i += 128) sA[i] = ws[off + i];
  __syncthreads();
#endif
}

// ---------------------------------------------------------------- setup ----
__global__ void setup_kernel(const float* __restrict__ A, const float* __restrict__ Bm,
                             const float* __restrict__ Pi, float* __restrict__ ws,
                             float* __restrict__ out) {
  int t = threadIdx.x; // 128 threads
  // softmax A over axis c; one (d,j,g) column per thread (128 columns)
  {
    int d = t >> 4, j = (t >> 2) & 3, g = t & 3;
    float v[8]; float mx = -1e30f;
    for (int c = 0; c < 8; ++c) { v[c] = A[c*128 + d*16 + j*4 + g]; mx = fmaxf(mx, v[c]); }
    float s = 0.f;
    for (int c = 0; c < 8; ++c) { v[c] = __expf(v[c]-mx); s += v[c]; }
    float inv = 1.f/s;
    for (int c = 0; c < 8; ++c) {
      float p = v[c]*inv;
      ws[WS_SMA_DC + ((j*4+g)*8 + d)*8 + c] = p;
      ws[WS_SMA_CD + ((j*4+g)*8 + c)*8 + d] = p;
    }
  }
  // softmax B_emit over axis m; one (c,g) row per thread (32 rows)
  if (t < 32) {
    int c = t >> 2, g = t & 3;
    float mx = -1e30f;
    for (int m = 0; m < MEMIT; ++m) mx = fmaxf(mx, Bm[c*400 + m*4 + g]);
    float s = 0.f;
    for (int m = 0; m < MEMIT; ++m) s += __expf(Bm[c*400 + m*4 + g]-mx);
    float inv = 1.f/s;
    for (int m = 0; m < MEMIT; ++m)
      ws[WS_SMBT + m*32 + g*8 + c] = __expf(Bm[c*400 + m*4 + g]-mx)*inv;
  }
  // softmax Pi over c; init the 8 root priors
  if (t < 4) {
    int g = t;
    float v[8]; float mx = -1e30f;
    for (int c = 0; c < 8; ++c) { v[c] = Pi[c*4+g]; mx = fmaxf(mx, v[c]); }
    float s = 0.f;
    for (int c = 0; c < 8; ++c) { v[c] = __expf(v[c]-mx); s += v[c]; }
    float inv = 1.f/s;
    for (int r = 0; r < 8; ++r)
      for (int c = 0; c < 8; ++c)
        ws[WS_PRIOR + r*32 + g*8 + c] = v[c]*inv;
  }
  if (t < 32) out[t] = 0.f; // atomic accumulation target
}

// ----------------------------------------------------- top-down (prior) ----
// Per wave: 16 parents. WMMA: M=16 parents, K=4 (d split {0..3},{4..7}),
// N=16 = (j-pair)x(8 states). 16 WMMAs -> 64 children x 32 (g,c) priors.
__global__ void down_kernel(float* __restrict__ ws, int n_pa, int off_pa, int off_ch) {
  __shared__ float sA[1024];
  load_table_lds(sA, ws, WS_SMA_DC);
  float* prior = ws + WS_PRIOR;
  int lane = threadIdx.x & 31, wave = threadIdx.x >> 5;
  int h = lane >> 4, nm = lane & 15;             // A row m = nm ; B/D col n = nm
  int jp = nm >> 3, cc = nm & 7;                 // N packing: n = jp*8 + c
  int p_base = blockIdx.x*64 + wave*16;
  bool full = (p_base + 16 <= n_pa);             // wave-uniform store guard
  __builtin_prefetch(&prior[(off_pa + p_base + 64)*32], 0, 0);
  for (int g = 0; g < 4; ++g) {
    int pa = (off_pa + p_base + nm)*32 + g*8;    // + d
    v2f a0 = *(const v2f*)(prior + pa + 2*h);        // d = 2h, 2h+1
    v2f a1 = *(const v2f*)(prior + pa + 4 + 2*h);    // d = 4+2h, 5+2h
    for (int jbase = 0; jbase < 4; jbase += 2) {
      int j = jbase + jp;
      int bi = ((j*4+g)*8)*8 + cc;               // sA[((j,g)*8 + d)*8 + c]
      v2f b0, b1;                                // B[k,n] = P(c|d=dbase+k, j(n), g)
      b0.x = sA[bi + (2*h + 0)*8];  b0.y = sA[bi + (2*h + 1)*8];
      b1.x = sA[bi + (2*h + 4)*8];  b1.y = sA[bi + (2*h + 5)*8];
      v8f acc = {0.f,0.f,0.f,0.f,0.f,0.f,0.f,0.f};
      acc = __builtin_amdgcn_wmma_f32_16x16x4_f32(false, a0, false, b0, (short)0, acc, false, false);
      acc = __builtin_amdgcn_wmma_f32_16x16x4_f32(false, a1, false, b1, (short)0, acc, false, false);
      if (full) {
#pragma unroll
        for (int r = 0; r < 8; ++r)              // D[r+8h, n] -> child prior
          prior[(off_ch + (p_base + r + 8*h)*4 + j)*32 + g*8 + cc] = acc[r];
      } else {
#pragma unroll
        for (int r = 0; r < 8; ++r) {
          int p = p_base + r + 8*h;
          if (p < n_pa) prior[(off_ch + p*4 + j)*32 + g*8 + cc] = acc[r];
        }
      }
    }
  }
}

// ------------------------------------------------------------- leaves ------
__global__ void leaf_kernel(const int* __restrict__ x, float* __restrict__ ws,
                            float* __restrict__ out) {
  __shared__ float accS[32];
  if (threadIdx.x < 32) accS[threadIdx.x] = 0.f;
  __syncthreads();
  int lane = threadIdx.x & 31, wave = threadIdx.x >> 5;   // lane = g*8 + c
  int node_in = blockIdx.x*8 + wave;
  int node_abs = 174760 + node_in;                        // OFFS[8]
  int xv = x[node_abs];
  float v = ws[WS_PRIOR + node_abs*32 + lane] * ws[WS_SMBT + xv*32 + lane];
  float nu = v;                                           // sum over c (xor 1,2,4)
  nu += __shfl_xor(nu, 1);
  nu += __shfl_xor(nu, 2);
  nu += __shfl_xor(nu, 4);
  ws[WS_BETA0 + node_in*32 + lane] = v/nu;
  if ((lane & 7) == 0) atomicAdd(&accS[(node_in >> 16)*4 + (lane >> 3)], __logf(nu));
  __syncthreads();
  if (threadIdx.x < 32) atomicAdd(&out[threadIdx.x], accS[threadIdx.x]);
}

// ---------------------------------------------------- bottom-up (beta) -----
// Per wave: 16 parents. Per (g,j): WMMA M=16 parents, K=c (split 2x4),
// N = d (8 valid cols, top 8 zeroed). Product over j, times beta0, normalize.
__global__ void up_kernel(const int* __restrict__ x, float* __restrict__ ws,
                          float* __restrict__ out,
                          const float* __restrict__ beta_in, float* __restrict__ beta_out,
                          int n_pa, int off_pa, int off_ch, int shift) {
  __shared__ float sA[1024];
  __shared__ float accS[32];
  if (threadIdx.x < 32) accS[threadIdx.x] = 0.f;
  load_table_lds(sA, ws, WS_SMA_CD);
  const float* prior = ws + WS_PRIOR;
  const float* smBt  = ws + WS_SMBT;
  int lane = threadIdx.x & 31, wave = threadIdx.x >> 5;
  int h = lane >> 4, nm = lane & 15;
  int dd = nm & 7;                                // clamped d for safe addressing
  float dmask = (nm < 8) ? 1.f : 0.f;             // zero invalid N columns
  int p_base = blockIdx.x*64 + wave*16;
  bool full = (p_base + 16 <= n_pa);              // wave-uniform guard
  for (int g = 0; g < 4; ++g) {
    v8f uv[4];
#pragma unroll
    for (int j = 0; j < 4; ++j) {
      int child_in = (p_base + nm)*4 + j;
      int cb = child_in*32 + g*8;                 // beta (per-level buffer)
      int ca = (off_ch + child_in)*32 + g*8;      // prior (absolute)
      v2f bb0 = *(const v2f*)(beta_in + cb + 2*h);      // c = 2h, 2h+1
      v2f pp0 = *(const v2f*)(prior   + ca + 2*h);
      v2f bb1 = *(const v2f*)(beta_in + cb + 4 + 2*h);  // c = 4+2h, 5+2h
      v2f pp1 = *(const v2f*)(prior   + ca + 4 + 2*h);
      v2f a0 = bb0 / pp0;                         // A[m,k] = ratio[child][c][g]
      v2f a1 = bb1 / pp1;
      int bi = ((j*4+g)*8)*8 + dd;                // sA[((j,g)*8 + c)*8 + d]
      v2f b0, b1;                                 // B[k,n] = P(c=cbase+k | d=n, j, g)
      b0.x = dmask * sA[bi + (2*h+0)*8];  b0.y = dmask * sA[bi + (2*h+1)*8];
      b1.x = dmask * sA[bi + (2*h+4)*8];  b1.y = dmask * sA[bi + (2*h+5)*8];
      v8f acc = {0.f,0.f,0.f,0.f,0.f,0.f,0.f,0.f};
      acc = __builtin_amdgcn_wmma_f32_16x16x4_f32(false, a0, false, b0, (short)0, acc, false, false);
      acc = __builtin_amdgcn_wmma_f32_16x16x4_f32(false, a1, false, b1, (short)0, acc, false, false);
      uv[j] = acc;
    }
    v8f prod = uv[0]*uv[1]*uv[2]*uv[3];           // product over children j
#pragma unroll
    for (int r = 0; r < 8; ++r) {
      int p = p_base + r + 8*h;
      int pabs = off_pa + p;
      int xv = x[pabs];
      float t = dmask * prod[r] * prior[pabs*32 + g*8 + dd] * smBt[xv*32 + g*8 + dd];
      float nu = t;                               // sum over d (xor 1,2,4 in 8-group)
      nu += __shfl_xor(nu, 1);
      nu += __shfl_xor(nu, 2);
      nu += __shfl_xor(nu, 4);
      bool ok = full || (p < n_pa);
      if (ok && nm < 8) beta_out[p*32 + g*8 + nm] = t/nu;
      if (ok && nm == 0) atomicAdd(&accS[(p >> shift)*4 + g], __logf(nu));
    }
  }
  __syncthreads();
  if (threadIdx.x < 32) atomicAdd(&out[threadIdx.x], accS[threadIdx.x]);
}

// ------------------------------------------------------------- launch ------
extern "C" void kernel_launch(void* const* d_in, const int* in_sizes, int n_in,
                              void* d_out, int out_size, void* d_ws, size_t ws_size,
                              hipStream_t stream) {
  (void)in_sizes; (void)n_in; (void)out_size; (void)ws_size;
  const int*   x  = (const int*)d_in[0];
  const float* A  = (const float*)d_in[1];
  const float* Bm = (const float*)d_in[2];
  const float* Pi = (const float*)d_in[3];
  float* out = (float*)d_out;
  float* ws  = (float*)d_ws;

  long counts[9], offs[10];
  offs[0] = 0;
  for (int l = 0; l <= 8; ++l) { counts[l] = 8L << (2*l); offs[l+1] = offs[l] + counts[l]; }

  setup_kernel<<<1, 128, 0, stream>>>(A, Bm, Pi, ws, out);

  for (int l = 1; l <= 8; ++l) {                         // top-down priors
    int n_pa = (int)counts[l-1];
    int blocks = (n_pa + 63)/64;
    down_kernel<<<blocks, 128, 0, stream>>>(ws, n_pa, (int)offs[l-1], (int)offs[l]);
  }

  leaf_kernel<<<(int)(counts[8]/8), 256, 0, stream>>>(x, ws, out);

  float* bufs[2] = { ws + WS_BETA0, ws + WS_BETA1 };     // ping-pong per level
  for (int l = 8; l >= 1; --l) {                         // bottom-up betas + ll
    int n_pa = (int)counts[l-1];
    int blocks = (n_pa + 63)/64;
    up_kernel<<<blocks, 128, 0, stream>>>(x, ws, out,
        bufs[l & 1], bufs[(l-1) & 1], n_pa, (int)offs[l-1], (int)offs[l], 2*(l-1));
  }
}